// SingleStreamBlock_62251255988774
// MI455X (gfx1250) — compile-verified
//
#include <hip/hip_runtime.h>
#include <math.h>
#include <stdint.h>

// ---------------- problem constants ----------------
#define S_    4096
#define H_    3072
#define NH_   24
#define HD_   128
#define MLP_  12288
#define H3_   9216      // 3*H
#define N1_   21504     // 3*H + MLP
#define HALF_ 64
#define EPS_  1e-6f

typedef __attribute__((ext_vector_type(2))) float v2f;
typedef __attribute__((ext_vector_type(4))) float v4f;
typedef __attribute__((ext_vector_type(8))) float v8f;

// GCC-style vector for async-LDS builtin parameter type
typedef int v4i_ __attribute__((vector_size(16)));

__device__ __forceinline__ float siluf(float x) { return x / (1.0f + __expf(-x)); }

__device__ __forceinline__ float gelu_tanh(float x) {
    const float c0 = 0.7978845608028654f;   // sqrt(2/pi)
    float x3 = x * x * x;
    float t  = tanhf(c0 * (x + 0.044715f * x3));
    return 0.5f * x * (1.0f + t);
}

// ---------------- CDNA5 async global->LDS copy (ASYNCcnt path) ----------------
// 16B per lane, written directly into LDS by the async engine (no VGPR staging).
__device__ __forceinline__ void async_ld16(const float* g, float* l) {
#if __has_builtin(__builtin_amdgcn_global_load_async_to_lds_b128)
    __builtin_amdgcn_global_load_async_to_lds_b128(
        (__attribute__((address_space(1))) v4i_*)(uintptr_t)g,
        (__attribute__((address_space(3))) v4i_*)(uint32_t)(uintptr_t)l,
        0, 0);
#else
    unsigned int loff = (unsigned int)(uintptr_t)l;   // low 32 bits = LDS byte address
    asm volatile("global_load_async_to_lds_b128 %0, %1, off"
                 :: "v"(loff), "v"(g) : "memory");
#endif
}

__device__ __forceinline__ void wait_async0() {
#if __has_builtin(__builtin_amdgcn_s_wait_asynccnt)
    __builtin_amdgcn_s_wait_asynccnt(0);
#else
    asm volatile("s_wait_asynccnt 0" ::: "memory");
#endif
}

// ================================================================
// Kernel 1: emb = silu(vec) @ norm_lin_w + norm_lin_b   (1x3072 @ 3072x9216)
// ================================================================
__global__ __launch_bounds__(256)
void k_emb(const float* __restrict__ vec, const float* __restrict__ W,
           const float* __restrict__ b, float* __restrict__ emb) {
    __shared__ float sv[256];
    const int j = blockIdx.x * 256 + threadIdx.x;   // 0..9215
    float acc = 0.0f;
    for (int kb = 0; kb < H_; kb += 256) {
        __syncthreads();
        sv[threadIdx.x] = siluf(vec[kb + threadIdx.x]);
        __syncthreads();
#pragma unroll 8
        for (int kk = 0; kk < 256; ++kk)
            acc = fmaf(sv[kk], W[(size_t)(kb + kk) * H3_ + j], acc);
    }
    emb[j] = acc + b[j];
}

// ================================================================
// Tiled fp32 WMMA GEMM  C[M,N] = A[M,K] @ B[K,N] + bias
//   MODE 0: A-load fuses adaLN modulation  a' = a*(1+scale[k]) + shift[k]
//   MODE 1: plain A (async-copied to LDS)
//   MODE 2: A = [o_proj (K<H) | gelu(h_mlp) (K>=H)], epilogue:
//           out = residual + gate[n]*(acc + bias[n])
// B tiles are always pure copies -> async global->LDS (ASYNCcnt).
// Block: 256 threads = 8 waves (2x4).  Tile 128x128, BK=32.
// Per wave: 64x32 output = 4x2 tiles of 16x16 fp32 WMMA accumulators.
// ================================================================
#define BM_ 128
#define BN_ 128
#define BK_ 32

template <int MODE>
__global__ __launch_bounds__(256)
void k_gemm(const float* __restrict__ A, int lda,
            const float* __restrict__ Bm, int ldb,
            const float* __restrict__ bias,
            float* __restrict__ C, int ldc,
            int Kdim,
            const float* __restrict__ emb,          // MODE0: shift=emb, scale=emb+H; MODE2: gate=emb+2H
            const float* __restrict__ A2, int lda2, // MODE2: mlp part of h (row stride N1)
            const float* __restrict__ residual)     // MODE2
{
    __shared__ __align__(16) float sA[BM_ * BK_];
    __shared__ __align__(16) float sB[BK_ * BN_];

    const int tid  = threadIdx.x;
    const int lane = tid & 31;
    const int wave = tid >> 5;       // 0..7
    const int wr   = wave >> 2;      // 0..1  (row of wave grid)
    const int wc   = wave & 3;       // 0..3  (col of wave grid)
    const int m0   = blockIdx.y * BM_;
    const int n0   = blockIdx.x * BN_;

    const int l    = lane & 15;            // M (A) or N (B/C) index within 16
    const int khi  = (lane >> 4) << 1;     // lanes 16-31 hold K=+2,+3

    v8f acc[4][2];
#pragma unroll
    for (int mi = 0; mi < 4; ++mi)
#pragma unroll
        for (int ni = 0; ni < 2; ++ni) {
            v8f z = {0.f, 0.f, 0.f, 0.f, 0.f, 0.f, 0.f, 0.f};
            acc[mi][ni] = z;
        }

    for (int kt = 0; kt < Kdim; kt += BK_) {
        __syncthreads();   // previous tile fully consumed before overwriting LDS

        // ---- B tile (32 x 128): async global->LDS, 1024 x b128, 4 per lane ----
#pragma unroll
        for (int it = 0; it < 4; ++it) {
            int i = it * 256 + tid;
            int r = i >> 5;             // 0..31
            int c = (i & 31) << 2;      // 0..124
            async_ld16(Bm + (size_t)(kt + r) * ldb + n0 + c, sB + r * BN_ + c);
        }

        // ---- A tile (128 x 32) ----
        if (MODE == 1) {
            // pure copy -> async path
#pragma unroll
            for (int it = 0; it < 4; ++it) {
                int i = it * 256 + tid;
                int r = i >> 3;             // 0..127
                int c = (i & 7) << 2;       // 0,4,...,28
                async_ld16(A + (size_t)(m0 + r) * lda + kt + c, sA + r * BK_ + c);
            }
        } else {
#pragma unroll
            for (int it = 0; it < 4; ++it) {
                int i = it * 256 + tid;
                int r = i >> 3;             // 0..127
                int c = (i & 7) << 2;       // 0,4,...,28
                int gk = kt + c;
                v4f av;
                if (MODE == 2) {
                    if (gk < H_) {
                        av = *(const v4f*)(A + (size_t)(m0 + r) * lda + gk);
                    } else {
                        av = *(const v4f*)(A2 + (size_t)(m0 + r) * lda2 + (gk - H_));
                        av.x = gelu_tanh(av.x); av.y = gelu_tanh(av.y);
                        av.z = gelu_tanh(av.z); av.w = gelu_tanh(av.w);
                    }
                } else { // MODE 0: adaLN modulation
                    av = *(const v4f*)(A + (size_t)(m0 + r) * lda + gk);
                    v4f sh = *(const v4f*)(emb + gk);         // shift
                    v4f sc = *(const v4f*)(emb + H_ + gk);    // scale
                    av.x = fmaf(av.x, 1.0f + sc.x, sh.x);
                    av.y = fmaf(av.y, 1.0f + sc.y, sh.y);
                    av.z = fmaf(av.z, 1.0f + sc.z, sh.z);
                    av.w = fmaf(av.w, 1.0f + sc.w, sh.w);
                }
                *(v4f*)(sA + r * BK_ + c) = av;
            }
        }

        // ---- prefetch next K tile into L2 while async copies land ----
        if (kt + BK_ < Kdim) {
            int gk2 = kt + BK_;
            const float* pa;
            if (MODE == 2 && gk2 >= H_)
                pa = A2 + (size_t)(m0 + (tid & 127)) * lda2 + (gk2 - H_);
            else
                pa = A + (size_t)(m0 + (tid & 127)) * lda + gk2;
            __builtin_prefetch(pa, 0, 0);
            __builtin_prefetch(Bm + (size_t)(gk2 + (tid & 31)) * ldb + n0 + ((tid >> 5) << 4), 0, 0);
        }

        wait_async0();       // fence this wave's async LDS writes
        __syncthreads();     // publish LDS tile to all waves

        // ---- compute: 8 WMMA k-steps of 4 ----
#pragma unroll
        for (int k0 = 0; k0 < BK_; k0 += 4) {
            const int kk = k0 + khi;   // this lane's K pair base (even)
            // B fragments (4x16): lanes 0-15 N=l hold K=k0,k0+1; lanes 16-31 K=k0+2,k0+3
            v2f bf[2];
#pragma unroll
            for (int ni = 0; ni < 2; ++ni) {
                int col = wc * 32 + ni * 16 + l;
                bf[ni].x = sB[kk * BN_ + col];
                bf[ni].y = sB[(kk + 1) * BN_ + col];
            }
#pragma unroll
            for (int mi = 0; mi < 4; ++mi) {
                int row = wr * 64 + mi * 16 + l;
                // A fragment (16x4): contiguous K pair -> aligned float2 LDS load
                v2f af = *(const v2f*)(sA + row * BK_ + kk);
                acc[mi][0] = __builtin_amdgcn_wmma_f32_16x16x4_f32(
                    false, af, false, bf[0], (short)0, acc[mi][0], false, false);
                acc[mi][1] = __builtin_amdgcn_wmma_f32_16x16x4_f32(
                    false, af, false, bf[1], (short)0, acc[mi][1], false, false);
            }
        }
    }

    // ---- epilogue: C 16x16 layout: VGPR r -> M=r (lanes 0-15) / M=8+r (lanes 16-31), N=l ----
    const int nlo = lane & 15;
    const int mhi = (lane >> 4) * 8;
#pragma unroll
    for (int mi = 0; mi < 4; ++mi) {
#pragma unroll
        for (int ni = 0; ni < 2; ++ni) {
            int row0 = m0 + wr * 64 + mi * 16 + mhi;
            int col  = n0 + wc * 32 + ni * 16 + nlo;
            float bc = bias[col];
            float g  = (MODE == 2) ? emb[2 * H_ + col] : 0.0f;
#pragma unroll
            for (int r = 0; r < 8; ++r) {
                size_t idx = (size_t)(row0 + r) * ldc + col;
                float v = acc[mi][ni][r] + bc;
                if (MODE == 2) v = residual[idx] + g * v;
                C[idx] = v;
            }
        }
    }
}

// ================================================================
// Kernel 3: per-token head-attention (24x24) with RMSNorm + RoPE
// One 128-thread block per token.
// ================================================================
__global__ __launch_bounds__(128)
void k_attn(const float* __restrict__ h,          // S x N1  (qkv in first 9216 cols)
            const float* __restrict__ fc, const float* __restrict__ fs,  // S x 64
            const float* __restrict__ qs, const float* __restrict__ ks,  // 128
            float* __restrict__ o)                 // S x H
{
    __shared__ float kb[NH_ * HD_];   // roped+normed K
    __shared__ float vb[NH_ * HD_];   // V
    __shared__ float qb[HD_];         // roped+normed Q (current head)
    __shared__ float tmp[HD_];
    __shared__ float red[HD_];
    __shared__ float sc[NH_];

    const int t = blockIdx.x;
    const int d = threadIdx.x;        // 0..127
    const float* base = h + (size_t)t * N1_;

    const float cfc = fc[t * HALF_ + (d & 63)];
    const float cfs = fs[t * HALF_ + (d & 63)];

    // load V
    for (int hh = 0; hh < NH_; ++hh)
        vb[hh * HD_ + d] = base[2 * H_ + hh * HD_ + d];

    // K: rmsnorm + rope per head
    for (int hh = 0; hh < NH_; ++hh) {
        float kv = base[H_ + hh * HD_ + d];
        red[d] = kv * kv; __syncthreads();
        for (int off = 64; off > 0; off >>= 1) {
            if (d < off) red[d] += red[d + off];
            __syncthreads();
        }
        float rr = rsqrtf(red[0] * (1.0f / HD_) + EPS_);
        tmp[d] = kv * rr * ks[d]; __syncthreads();
        if (d < HALF_) {
            float x0 = tmp[2 * d], x1 = tmp[2 * d + 1];
            kb[hh * HD_ + d]         = cfc * x0 - cfs * x1;
            kb[hh * HD_ + HALF_ + d] = cfs * x0 + cfc * x1;
        }
        __syncthreads();
    }

    // Q heads: rmsnorm + rope + scores + softmax + AV
    for (int i = 0; i < NH_; ++i) {
        float qv = base[i * HD_ + d];
        red[d] = qv * qv; __syncthreads();
        for (int off = 64; off > 0; off >>= 1) {
            if (d < off) red[d] += red[d + off];
            __syncthreads();
        }
        float rr = rsqrtf(red[0] * (1.0f / HD_) + EPS_);
        tmp[d] = qv * rr * qs[d]; __syncthreads();
        if (d < HALF_) {
            float x0 = tmp[2 * d], x1 = tmp[2 * d + 1];
            qb[d]         = cfc * x0 - cfs * x1;
            qb[HALF_ + d] = cfs * x0 + cfc * x1;
        }
        __syncthreads();
        if (d < NH_) {
            float s = 0.0f;
#pragma unroll 8
            for (int e = 0; e < HD_; ++e)
                s = fmaf(qb[e], kb[d * HD_ + e], s);
            sc[d] = s * 0.08838834764831845f;   // 128^-0.5
        }
        __syncthreads();
        // softmax over 24 (computed redundantly by all lanes) + AV
        float mx = -1e30f;
#pragma unroll
        for (int j = 0; j < NH_; ++j) mx = fmaxf(mx, sc[j]);
        float den = 0.0f, ov = 0.0f;
#pragma unroll
        for (int j = 0; j < NH_; ++j) {
            float e = __expf(sc[j] - mx);
            den += e;
            ov = fmaf(e, vb[j * HD_ + d], ov);
        }
        o[(size_t)t * H_ + i * HD_ + d] = ov / den;
        __syncthreads();
    }
}

// ================================================================
// Launcher
// ================================================================
extern "C" void kernel_launch(void* const* d_in, const int* in_sizes, int n_in,
                              void* d_out, int out_size, void* d_ws, size_t ws_size,
                              hipStream_t stream) {
    (void)in_sizes; (void)n_in; (void)out_size; (void)ws_size;
    const float* x    = (const float*)d_in[0];
    const float* vec  = (const float*)d_in[1];
    const float* fc   = (const float*)d_in[2];
    const float* fs   = (const float*)d_in[3];
    const float* nlw  = (const float*)d_in[4];
    const float* nlb  = (const float*)d_in[5];
    const float* l1w  = (const float*)d_in[6];
    const float* l1b  = (const float*)d_in[7];
    const float* qs   = (const float*)d_in[8];
    const float* ks   = (const float*)d_in[9];
    const float* pw   = (const float*)d_in[10];
    const float* pb   = (const float*)d_in[11];
    const float* pow_ = (const float*)d_in[12];
    const float* pob  = (const float*)d_in[13];

    float* ws    = (float*)d_ws;
    float* emb   = ws;                                  // 9216
    float* hbuf  = ws + 16384;                          // S*N1  (352 MB)
    float* attno = hbuf + (size_t)S_ * N1_;             // S*H
    float* oproj = attno + (size_t)S_ * H_;             // S*H
    float* out   = (float*)d_out;

    // 1) modulation vector
    k_emb<<<dim3(H3_ / 256), 256, 0, stream>>>(vec, nlw, nlb, emb);

    // 2) h = mod(x) @ linear1_w + b      (4096 x 3072 x 21504)
    k_gemm<0><<<dim3(N1_ / BN_, S_ / BM_), 256, 0, stream>>>(
        x, H_, l1w, N1_, l1b, hbuf, N1_, H_, emb, nullptr, 0, nullptr);

    // 3) per-token head attention
    k_attn<<<dim3(S_), 128, 0, stream>>>(hbuf, fc, fs, qs, ks, attno);

    // 4) o_proj = attn_o @ proj_w + proj_b   (4096 x 3072 x 3072)
    k_gemm<1><<<dim3(H_ / BN_, S_ / BM_), 256, 0, stream>>>(
        attno, H_, pw, H_, pb, oproj, H_, H_, nullptr, nullptr, 0, nullptr);

    // 5) out = residual + gate * ([o_proj | gelu(mlp)] @ proj_out_w + b)
    //    (4096 x 15360 x 3072), A2 = mlp slice of h
    k_gemm<2><<<dim3(H_ / BN_, S_ / BM_), 256, 0, stream>>>(
        oproj, H_, pow_, H_, pob, out, H_, H_ + MLP_, emb, hbuf + H3_, N1_, x);
}